// Equalize_73443940762375
// MI455X (gfx1250) — compile-verified
//
#include <hip/hip_runtime.h>
#include <stdint.h>

#define IMGS 64
#define HW   (512*512)            // pixels per channel
#define CH   3
#define PER_IMG (HW*CH)           // 786432 ints per image
#define V4_PER_IMG (PER_IMG/4)    // 196608 int4 per image
#define BPI 16                    // blocks per image (hist & apply)
#define TPB 256

// ---------------- workspace layout ----------------
// [0, 64*768*4)                       : unsigned ghist[64][3][256]
// [64*768*4, 64*768*4 + 192*256*4)    : float lut[192][256]

__global__ void zero_hist(unsigned* __restrict__ g, int n) {
    int i = blockIdx.x * blockDim.x + threadIdx.x;
    if (i < n) g[i] = 0u;
}

__global__ __launch_bounds__(TPB)
void hist_kernel(const int4* __restrict__ in, unsigned* __restrict__ ghist) {
    __shared__ unsigned hist[CH * 256];
    const int img = blockIdx.x / BPI;
    const int sub = blockIdx.x % BPI;

    for (int i = threadIdx.x; i < CH * 256; i += TPB) hist[i] = 0u;
    __syncthreads();

    const int4* base = in + (size_t)img * V4_PER_IMG;
    for (int t = sub * TPB + threadIdx.x; t < V4_PER_IMG; t += BPI * TPB) {
        int4 p = base[t];                 // global_load_b128
        int c0 = t % 3;
        int c1 = c0 + 1; if (c1 == 3) c1 = 0;
        int c2 = c1 + 1; if (c2 == 3) c2 = 0;
        atomicAdd(&hist[c0 * 256 + p.x], 1u);   // ds_add_u32 (no return)
        atomicAdd(&hist[c1 * 256 + p.y], 1u);
        atomicAdd(&hist[c2 * 256 + p.z], 1u);
        atomicAdd(&hist[c0 * 256 + p.w], 1u);
    }
    __syncthreads();

    unsigned* g = ghist + (size_t)img * (CH * 256);
    for (int i = threadIdx.x; i < CH * 256; i += TPB) {
        unsigned v = hist[i];
        if (v) atomicAdd(&g[i], v);       // global_atomic_add_u32
    }
}

// One block per (image, channel): 192 blocks x 256 threads (one thread per bin).
__global__ __launch_bounds__(256)
void lut_kernel(const unsigned* __restrict__ ghist, float* __restrict__ lut) {
    __shared__ unsigned sh[256];
    __shared__ int sidx[256];
    const int tid = threadIdx.x;
    const unsigned h = ghist[(size_t)blockIdx.x * 256 + tid];

    sh[tid]   = h;
    sidx[tid] = (h != 0u) ? tid : -1;
    __syncthreads();

    // max-reduce to find last nonzero bin index
    for (int s = 128; s > 0; s >>= 1) {
        if (tid < s) { int o = sidx[tid + s]; if (o > sidx[tid]) sidx[tid] = o; }
        __syncthreads();
    }
    const int last = sidx[0];                       // >= 0 (histogram sums to HW)
    const unsigned last_nonzero = sh[last];         // read before scan mutates sh
    const int step  = (int)(((unsigned)HW - last_nonzero) / 255u);
    const int sstep = (step > 0) ? step : 1;

    // Hillis-Steele inclusive scan over sh[256]
    for (int off = 1; off < 256; off <<= 1) {
        unsigned v = 0;
        if (tid >= off) v = sh[tid - off];
        __syncthreads();
        sh[tid] += v;
        __syncthreads();
    }
    const unsigned excl = sh[tid] - h;              // exclusive prefix sum

    float outv;
    if (step == 0) {
        outv = (float)tid;                          // identity LUT == "out = img"
    } else {
        unsigned q = (excl + (unsigned)(sstep >> 1)) / (unsigned)sstep;
        if (q > 255u) q = 255u;
        outv = (float)q;
    }
    lut[(size_t)blockIdx.x * 256 + tid] = outv;
}

__global__ __launch_bounds__(TPB)
void apply_kernel(const int4* __restrict__ in, const float* __restrict__ lut,
                  float4* __restrict__ out) {
    __shared__ float slut[CH * 256];                // 3 KB LUT for this image
    const int img = blockIdx.x / BPI;
    const int sub = blockIdx.x % BPI;

    // Wave 0 stages the 3KB LUT into LDS via the CDNA5 async global->LDS path:
    // 6 x (32 lanes x 16B) = 3072B, tracked by ASYNCcnt.
    if (threadIdx.x < 32) {
        unsigned lane = threadIdx.x;
        unsigned ldsbase = (unsigned)(size_t)(&slut[0]);   // low 32 bits = LDS offset
        unsigned long long gbase =
            (unsigned long long)(size_t)(lut + (size_t)img * (CH * 256));
        #pragma unroll
        for (int ch = 0; ch < 6; ++ch) {
            unsigned la = ldsbase + lane * 16u + (unsigned)ch * 512u;
            unsigned long long ga = gbase + (unsigned long long)(lane * 16u + ch * 512u);
            asm volatile("global_load_async_to_lds_b128 %0, %1, off"
                         :: "v"(la), "v"(ga) : "memory");
        }
        asm volatile("s_wait_asynccnt 0" ::: "memory");
    }
    __syncthreads();

    const int4* ibase = in  + (size_t)img * V4_PER_IMG;
    float4*     obase = out + (size_t)img * V4_PER_IMG;
    for (int t = sub * TPB + threadIdx.x; t < V4_PER_IMG; t += BPI * TPB) {
        int4 p = ibase[t];                          // global_load_b128
        int c0 = t % 3;
        int c1 = c0 + 1; if (c1 == 3) c1 = 0;
        int c2 = c1 + 1; if (c2 == 3) c2 = 0;
        float4 o;
        o.x = slut[c0 * 256 + p.x];                 // ds_load_b32 gathers
        o.y = slut[c1 * 256 + p.y];
        o.z = slut[c2 * 256 + p.z];
        o.w = slut[c0 * 256 + p.w];
        obase[t] = o;                               // global_store_b128
    }
}

extern "C" void kernel_launch(void* const* d_in, const int* in_sizes, int n_in,
                              void* d_out, int out_size, void* d_ws, size_t ws_size,
                              hipStream_t stream) {
    const int4* in = (const int4*)d_in[0];                 // [64,512,512,3] int32
    float* out = (float*)d_out;                            // uint8 ref -> float out

    unsigned* ghist = (unsigned*)d_ws;                     // 64*3*256 u32
    float* lut = (float*)((char*)d_ws + (size_t)IMGS * CH * 256 * sizeof(unsigned));

    const int nhist = IMGS * CH * 256;
    zero_hist<<<(nhist + 255) / 256, 256, 0, stream>>>(ghist, nhist);
    hist_kernel<<<IMGS * BPI, TPB, 0, stream>>>(in, ghist);
    lut_kernel<<<IMGS * CH, 256, 0, stream>>>(ghist, lut);
    apply_kernel<<<IMGS * BPI, TPB, 0, stream>>>(in, lut, (float4*)out);
}